// Sim3Normalizer_46222438039687
// MI455X (gfx1250) — compile-verified
//
#include <hip/hip_runtime.h>

// Sim3 normalizer: per-frame y-axis rotation + neck-centering + 1/s scaling.
// Memory-bound (AI ~0.6 flop/byte): stream 4 frames/block (26064 B = 1629 x b128)
// into LDS with gfx1250 async LDS-DMA, compute from LDS, coalesced b96 stores.

#define NLM     543
#define FPB     4                   // frames per block
#define FRAME_F (NLM * 3)           // 1629 floats per frame
#define TILE_F  (FPB * FRAME_F)     // 6516 floats per block tile
#define TILE_B  (TILE_F * 4)        // 26064 bytes (16B multiple)
#define CHUNKS  (TILE_B / 16)       // 1629 b128 async copies
#define NT      256

struct F3 { float x, y, z; };

__global__ __launch_bounds__(NT)
void sim3_normalize_kernel(const float* __restrict__ in, float* __restrict__ out) {
    __shared__ __align__(16) float lds[TILE_F];
    __shared__ float prm[FPB][6];   // neck.xyz, cos, sin, 1/s

    const int    tid   = threadIdx.x;
    const size_t fbase = (size_t)blockIdx.x * TILE_F;   // float offset of tile
    const char*  gsrc  = (const char*)in + fbase * 4;
    const unsigned lbase = (unsigned)(size_t)&lds[0];   // low 32 bits == LDS addr

    // ---- Stage 1: async-stream 4 frames into LDS (b128, ASYNCcnt-tracked) ----
    for (int i = tid; i < CHUNKS; i += NT) {
        unsigned long long ga = (unsigned long long)(gsrc + (size_t)i * 16);
        unsigned           la = lbase + (unsigned)i * 16u;
        asm volatile("global_load_async_to_lds_b128 %0, %1, off"
                     :: "v"(la), "v"(ga) : "memory");
    }
    asm volatile("s_wait_asynccnt 0" ::: "memory");
    __syncthreads();

    // ---- Stage 2: per-frame rotation parameters (threads 0..3) ----
    if (tid < FPB) {
        const float* f = &lds[tid * FRAME_F];
        const float nx = f[0],  ny = f[1],  nz = f[2];     // neck  (idx 0)
        const float lx = f[33], ly = f[34], lz = f[35];    // L_SH  (idx 11)
        const float rx = f[36], ry = f[37], rz = f[38];    // R_SH  (idx 12)
        const float svx = rx - lx, svy = ry - ly, svz = rz - lz;
        const float s   = sqrtf(svx*svx + svy*svy + svz*svz) + 1e-8f;
        const float nxz = sqrtf(svx*svx + svz*svz) + 1e-8f;
        float c  = fminf(fmaxf(svx / nxz, -1.0f), 1.0f);
        float sn = sqrtf(fmaxf(1.0f - c*c, 0.0f));
        if (fabsf(c) > 0.9999f) { c = 1.0f; sn = 0.0f; }   // identity snap
        prm[tid][0] = nx; prm[tid][1] = ny; prm[tid][2] = nz;
        prm[tid][3] = c;  prm[tid][4] = sn; prm[tid][5] = 1.0f / s;
    }
    __syncthreads();

    // ---- Stage 3: rotate + scale each landmark; coalesced 12B stores ----
    for (int f = 0; f < FPB; ++f) {
        const float nx  = prm[f][0], ny = prm[f][1], nz = prm[f][2];
        const float c   = prm[f][3], sn = prm[f][4], inv = prm[f][5];
        const float* src = &lds[f * FRAME_F];
        float*       dst = out + fbase + (size_t)f * FRAME_F;
        for (int n = tid; n < NLM; n += NT) {
            const float px = src[n*3 + 0] - nx;
            const float py = src[n*3 + 1] - ny;
            const float pz = src[n*3 + 2] - nz;
            F3 o;
            o.x = ( c * px + sn * pz) * inv;
            o.y =   py * inv;
            o.z = (-sn * px +  c * pz) * inv;
            *(F3*)(dst + n*3) = o;
        }
    }
}

extern "C" void kernel_launch(void* const* d_in, const int* in_sizes, int n_in,
                              void* d_out, int out_size, void* d_ws, size_t ws_size,
                              hipStream_t stream) {
    (void)n_in; (void)out_size; (void)d_ws; (void)ws_size;
    const float* in  = (const float*)d_in[0];
    float*       out = (float*)d_out;
    const int frames = in_sizes[0] / FRAME_F;   // 32768
    const int blocks = frames / FPB;            // 8192 (T is a multiple of 4)
    sim3_normalize_kernel<<<blocks, NT, 0, stream>>>(in, out);
}